// EngineeringGNN_26534307954693
// MI455X (gfx1250) — compile-verified
//
#include <hip/hip_runtime.h>

// ---------------------------------------------------------------------------
// EngineeringGNN on MI455X (gfx1250, wave32).
// fp32 throughout (matches reference dtype); dense 128x128 GEMMs use
// V_WMMA_F32_16X16X4_F32. Weight staging into LDS uses the Tensor Data Mover
// (TENSOR_LOAD_TO_LDS + s_wait_tensorcnt) when the builtin is available.
// Workload is memory-bound on the edge phase (~3.5 GB @ 23.3 TB/s), so fp32
// WMMA is the right precision choice.
// ---------------------------------------------------------------------------

typedef float v2f __attribute__((ext_vector_type(2)));
typedef float v8f __attribute__((ext_vector_type(8)));
typedef unsigned int v4u __attribute__((ext_vector_type(4)));
typedef int v4i __attribute__((ext_vector_type(4)));
typedef int v8i __attribute__((ext_vector_type(8)));

#define HDIM 128
#define HSQ  (HDIM * HDIM)

__device__ __forceinline__ float relu_f(float x) { return fmaxf(x, 0.0f); }

// ---------------------------------------------------------------------------
// Stage a 128x128 fp32 weight matrix from global into LDS.
// Preferred path: one TDM descriptor per block, issued by wave 0:
//   D# group0: count=1, lds_addr, global_addr(57b), type=2
//   D# group1: data_size=4B, tensor_dim0=128, tensor_dim1=128,
//              tile_dim0=128, tile_dim1=128, tensor_dim0_stride=128
// Completion: s_wait_tensorcnt 0 (issuing wave), then workgroup barrier.
// Fallback: cooperative vector loads.
// ---------------------------------------------------------------------------
__device__ __forceinline__ void stage_weights(const float* __restrict__ W,
                                              float* __restrict__ sW)
{
#if __has_builtin(__builtin_amdgcn_tensor_load_to_lds)
    if (threadIdx.x < 32) {
        // Low 32 bits of a flat shared pointer = LDS byte offset.
        const unsigned int lds_addr = (unsigned int)(size_t)sW;
        const unsigned long long ga = (unsigned long long)(size_t)W;
        v4u g0;
        g0[0] = 1u;                            // count=1, user mode, no gather
        g0[1] = lds_addr;                      // lds_addr [63:32]
        g0[2] = (unsigned int)ga;              // global_addr [95:64]
        g0[3] = (unsigned int)((ga >> 32) & 0x01FFFFFFull) | 0x80000000u; // addr[56:32] | type=2
        v8i g1;
        g1[0] = 0x00020000;                    // workgroup_mask=0, data_size=2 (4B)
        g1[1] = (HDIM << 16);                  // tensor_dim0 low16 in [63:48]
        g1[2] = (HDIM << 16);                  // tensor_dim1 low16 in [95:80]
        g1[3] = (HDIM << 16);                  // tile_dim0 in [127:112]
        g1[4] = HDIM;                          // tile_dim1 in [143:128]
        g1[5] = HDIM;                          // tensor_dim0_stride low32 [191:160]
        g1[6] = 0;
        g1[7] = 0;
        v4i z4; z4[0] = 0; z4[1] = 0; z4[2] = 0; z4[3] = 0;
#if defined(__clang_major__) && (__clang_major__ >= 23)
        v8i z8; z8[0]=0; z8[1]=0; z8[2]=0; z8[3]=0; z8[4]=0; z8[5]=0; z8[6]=0; z8[7]=0;
        __builtin_amdgcn_tensor_load_to_lds(g0, g1, z4, z4, z8, 0);
#else
        __builtin_amdgcn_tensor_load_to_lds(g0, g1, z4, z4, 0);
#endif
#if __has_builtin(__builtin_amdgcn_s_wait_tensorcnt)
        __builtin_amdgcn_s_wait_tensorcnt(0);
#else
        asm volatile("s_wait_tensorcnt 0x0" ::: "memory");
#endif
    }
    __syncthreads();
#else
    for (int i = threadIdx.x; i < HSQ; i += 256) sW[i] = W[i];
    __syncthreads();
#endif
}

// ---------------------------------------------------------------------------
// WMMA fragment mapping (ISA 7.12.2, wave32):
//   A 16x4  f32 : lane L holds row m = L&15, k = 4*kk + 2*(L>>4) + {0,1}
//   B 4x16  f32 : lane L holds col n = L&15 (+16*j), same k pair
//   C 16x16 f32 : acc vgpr p, lane L -> row m = p + 8*(L>>4), col n = L&15
// One wave computes a 16x128 output tile: 8 accumulators, 32 k-steps.
// ---------------------------------------------------------------------------

// Generic Y = epilogue( (X [+ X2]) @ W + b ) for 128x128 W staged in LDS.
// LNRES=false: Y = relu(Z@W + b)
// LNRES=true : Y = LayerNorm( resid + relu(Z@W + b) ) * gamma + beta
template <bool ADD2, bool LNRES>
__global__ __launch_bounds__(256) void gemm128_kernel(
    const float* __restrict__ X, const float* __restrict__ X2,
    const float* __restrict__ W, const float* __restrict__ bias,
    const float* __restrict__ resid, const float* __restrict__ gamma,
    const float* __restrict__ beta, float* __restrict__ Y, int rows)
{
    __shared__ float sW[HSQ];                       // 64 KB
    stage_weights(W, sW);

    const int wave = threadIdx.x >> 5;
    const int lane = threadIdx.x & 31;
    const int half = lane >> 4;
    const int lm   = lane & 15;
    const int r0   = blockIdx.x * 128 + wave * 16;
    if (r0 >= rows) return;

    const int rA = min(r0 + lm, rows - 1);
    const float* xrow  = X + (size_t)rA * HDIM;
    const float* x2row = ADD2 ? (X2 + (size_t)rA * HDIM) : nullptr;

    v8f acc[8] = {};
    #pragma unroll 4
    for (int kk = 0; kk < 32; ++kk) {
        const int k0 = kk * 4 + half * 2;
        v2f a = *(const v2f*)(xrow + k0);
        if (ADD2) {
            v2f a2 = *(const v2f*)(x2row + k0);
            a[0] += a2[0]; a[1] += a2[1];
        }
        #pragma unroll
        for (int j = 0; j < 8; ++j) {
            const int n = j * 16 + lm;
            v2f b;
            b[0] = sW[k0 * HDIM + n];
            b[1] = sW[(k0 + 1) * HDIM + n];
            acc[j] = __builtin_amdgcn_wmma_f32_16x16x4_f32(
                false, a, false, b, (short)0, acc[j], false, false);
        }
    }

    float bcol[8], gcol[8], becol[8];
    #pragma unroll
    for (int j = 0; j < 8; ++j) {
        const int n = j * 16 + lm;
        bcol[j] = bias[n];
        if (LNRES) { gcol[j] = gamma[n]; becol[j] = beta[n]; }
    }

    if (!LNRES) {
        #pragma unroll
        for (int p = 0; p < 8; ++p) {
            const int r = r0 + p + half * 8;
            if (r >= rows) continue;
            #pragma unroll
            for (int j = 0; j < 8; ++j)
                Y[(size_t)r * HDIM + j * 16 + lm] = relu_f(acc[j][p] + bcol[j]);
        }
    } else {
        #pragma unroll
        for (int p = 0; p < 8; ++p) {
            const int r  = r0 + p + half * 8;
            const int rc = min(r, rows - 1);
            float v[8], s1 = 0.0f, s2 = 0.0f;
            #pragma unroll
            for (int j = 0; j < 8; ++j) {
                float c = acc[j][p] + bcol[j];
                float t = resid[(size_t)rc * HDIM + j * 16 + lm] + relu_f(c);
                v[j] = t; s1 += t; s2 += t * t;
            }
            // reduce across the 16-lane half (row lives in one half-wave)
            #pragma unroll
            for (int m = 1; m <= 8; m <<= 1) {
                s1 += __shfl_xor(s1, m, 32);
                s2 += __shfl_xor(s2, m, 32);
            }
            const float mean = s1 * (1.0f / 128.0f);
            const float var  = s2 * (1.0f / 128.0f) - mean * mean;
            const float inv  = rsqrtf(var + 1e-5f);
            if (r >= rows) continue;
            #pragma unroll
            for (int j = 0; j < 8; ++j)
                Y[(size_t)r * HDIM + j * 16 + lm] =
                    (v[j] - mean) * inv * gcol[j] + becol[j];
        }
    }
}

// Fused encoder: Y = LN( relu(X@w1 + b1) @ W2 + b2 ).  KIN = 5 (nodes) or 6
// (edges).  A-fragments of the second GEMM computed on the fly (2 dot-KIN
// per k-step) -> no E x 128 intermediate in memory.
template <int KIN>
__global__ __launch_bounds__(256) void encode_kernel(
    const float* __restrict__ X, const float* __restrict__ w1,
    const float* __restrict__ b1, const float* __restrict__ W2,
    const float* __restrict__ b2, const float* __restrict__ gamma,
    const float* __restrict__ beta, float* __restrict__ Y, int rows)
{
    __shared__ float sW[HSQ];
    stage_weights(W2, sW);

    const int wave = threadIdx.x >> 5;
    const int lane = threadIdx.x & 31;
    const int half = lane >> 4;
    const int lm   = lane & 15;
    const int r0   = blockIdx.x * 128 + wave * 16;
    if (r0 >= rows) return;

    const int rA = min(r0 + lm, rows - 1);
    float xr[KIN];
    #pragma unroll
    for (int q = 0; q < KIN; ++q) xr[q] = X[(size_t)rA * KIN + q];

    v8f acc[8] = {};
    #pragma unroll 2
    for (int kk = 0; kk < 32; ++kk) {
        const int k0 = kk * 4 + half * 2;
        float a0 = b1[k0], a1 = b1[k0 + 1];
        #pragma unroll
        for (int q = 0; q < KIN; ++q) {
            a0 = fmaf(xr[q], w1[q * HDIM + k0], a0);
            a1 = fmaf(xr[q], w1[q * HDIM + k0 + 1], a1);
        }
        v2f a; a[0] = relu_f(a0); a[1] = relu_f(a1);
        #pragma unroll
        for (int j = 0; j < 8; ++j) {
            const int n = j * 16 + lm;
            v2f b;
            b[0] = sW[k0 * HDIM + n];
            b[1] = sW[(k0 + 1) * HDIM + n];
            acc[j] = __builtin_amdgcn_wmma_f32_16x16x4_f32(
                false, a, false, b, (short)0, acc[j], false, false);
        }
    }

    float bcol[8], gcol[8], becol[8];
    #pragma unroll
    for (int j = 0; j < 8; ++j) {
        const int n = j * 16 + lm;
        bcol[j] = b2[n]; gcol[j] = gamma[n]; becol[j] = beta[n];
    }

    #pragma unroll
    for (int p = 0; p < 8; ++p) {
        const int r = r0 + p + half * 8;
        float v[8], s1 = 0.0f, s2 = 0.0f;
        #pragma unroll
        for (int j = 0; j < 8; ++j) {
            float t = acc[j][p] + bcol[j];
            v[j] = t; s1 += t; s2 += t * t;
        }
        #pragma unroll
        for (int m = 1; m <= 8; m <<= 1) {
            s1 += __shfl_xor(s1, m, 32);
            s2 += __shfl_xor(s2, m, 32);
        }
        const float mean = s1 * (1.0f / 128.0f);
        const float var  = s2 * (1.0f / 128.0f) - mean * mean;
        const float inv  = rsqrtf(var + 1e-5f);
        if (r >= rows) continue;
        #pragma unroll
        for (int j = 0; j < 8; ++j)
            Y[(size_t)r * HDIM + j * 16 + lm] =
                (v[j] - mean) * inv * gcol[j] + becol[j];
    }
}

// agg[dst] += relu(h[src] + ea)  -- one wave per edge, float4 per lane.
__global__ __launch_bounds__(256) void scatter_kernel(
    const float* __restrict__ h, const float* __restrict__ ea,
    const int* __restrict__ ei, float* __restrict__ agg, int E)
{
    const long tid = (long)blockIdx.x * 256 + threadIdx.x;
    const int e = (int)(tid >> 5);
    if (e >= E) return;
    const int c = (int)(tid & 31) * 4;
    const int s = ei[e];
    const int d = ei[E + e];
    const float4 hv = *(const float4*)(h + (size_t)s * HDIM + c);
    const float4 ev = *(const float4*)(ea + (size_t)e * HDIM + c);
    float* ap = agg + (size_t)d * HDIM + c;
    unsafeAtomicAdd(ap + 0, fmaxf(hv.x + ev.x, 0.0f));
    unsafeAtomicAdd(ap + 1, fmaxf(hv.y + ev.y, 0.0f));
    unsafeAtomicAdd(ap + 2, fmaxf(hv.z + ev.z, 0.0f));
    unsafeAtomicAdd(ap + 3, fmaxf(hv.w + ev.w, 0.0f));
}

__global__ __launch_bounds__(256) void zero4_kernel(float* __restrict__ p, long n4)
{
    const long i = (long)blockIdx.x * 256 + threadIdx.x;
    if (i < n4) ((float4*)p)[i] = make_float4(0.f, 0.f, 0.f, 0.f);
}

// Pack the two head first-layers side by side: Wcat[128][128], bcat[128].
__global__ __launch_bounds__(256) void pack_heads_kernel(
    const float* __restrict__ dh_w1, const float* __restrict__ dh_b1,
    const float* __restrict__ sh_w1, const float* __restrict__ sh_b1,
    float* __restrict__ wcat, float* __restrict__ bcat)
{
    const int t = blockIdx.x * 256 + threadIdx.x;
    if (t < HSQ) {
        const int r = t >> 7, c = t & 127;
        wcat[t] = (c < 64) ? dh_w1[r * 64 + c] : sh_w1[r * 64 + (c - 64)];
    } else if (t < HSQ + HDIM) {
        const int c = t - HSQ;
        bcat[c] = (c < 64) ? dh_b1[c] : sh_b1[c - 64];
    }
}

// Per-node 64->3 / 64->1 head tails + clip/exp + output assembly.
__global__ __launch_bounds__(256) void finalize_kernel(
    const float* __restrict__ thead, const float* __restrict__ dh_w2,
    const float* __restrict__ dh_b2, const float* __restrict__ sh_w2,
    const float* __restrict__ sh_b2, const float* __restrict__ log_disp,
    float* __restrict__ out, int N)
{
    const int i = blockIdx.x * 256 + threadIdx.x;
    if (i >= N) return;
    const float* tr = thead + (size_t)i * HDIM;
    float u0 = dh_b2[0], u1 = dh_b2[1], u2 = dh_b2[2];
    float ls = sh_b2[0];
    #pragma unroll 8
    for (int c = 0; c < 64; ++c) {
        const float tv = tr[c];
        u0 = fmaf(tv, dh_w2[c * 3 + 0], u0);
        u1 = fmaf(tv, dh_w2[c * 3 + 1], u1);
        u2 = fmaf(tv, dh_w2[c * 3 + 2], u2);
        ls = fmaf(tr[64 + c], sh_w2[c], ls);
    }
    ls = fminf(fmaxf(ls, 0.0f), 30.0f);
    const float s = expf(ls);
    const float d = 0.001f + log1pf(expf(log_disp[0]));   // MIN_DISP + softplus
    out[(size_t)3 * i + 0] = u0 * d;
    out[(size_t)3 * i + 1] = u1 * d;
    out[(size_t)3 * i + 2] = u2 * d;
    out[(size_t)3 * N + i] = s;
    out[(size_t)4 * N + i] = ls;
    out[(size_t)5 * N + 1 + i] = 2.5e8f / (s + 1e-8f);
    if (i == 0) out[(size_t)5 * N] = d;
}

// ---------------------------------------------------------------------------

extern "C" void kernel_launch(void* const* d_in, const int* in_sizes, int n_in,
                              void* d_out, int out_size, void* d_ws, size_t ws_size,
                              hipStream_t stream)
{
    const float* x         = (const float*)d_in[0];
    const float* edge_attr = (const float*)d_in[1];
    const int*   ei        = (const int*)d_in[2];
    const float* ne_w1 = (const float*)d_in[3];
    const float* ne_b1 = (const float*)d_in[4];
    const float* ne_w2 = (const float*)d_in[5];
    const float* ne_b2 = (const float*)d_in[6];
    const float* ne_g  = (const float*)d_in[7];
    const float* ne_be = (const float*)d_in[8];
    const float* ee_w1 = (const float*)d_in[9];
    const float* ee_b1 = (const float*)d_in[10];
    const float* ee_w2 = (const float*)d_in[11];
    const float* ee_b2 = (const float*)d_in[12];
    const float* ee_g  = (const float*)d_in[13];
    const float* ee_be = (const float*)d_in[14];
    const float* conv_w1 = (const float*)d_in[15];
    const float* conv_b1 = (const float*)d_in[16];
    const float* conv_w2 = (const float*)d_in[17];
    const float* conv_b2 = (const float*)d_in[18];
    const float* pn_g    = (const float*)d_in[19];
    const float* pn_be   = (const float*)d_in[20];
    const float* dh_w1 = (const float*)d_in[21];
    const float* dh_b1 = (const float*)d_in[22];
    const float* dh_w2 = (const float*)d_in[23];
    const float* dh_b2 = (const float*)d_in[24];
    const float* sh_w1 = (const float*)d_in[25];
    const float* sh_b1 = (const float*)d_in[26];
    const float* sh_w2 = (const float*)d_in[27];
    const float* sh_b2 = (const float*)d_in[28];
    const float* log_disp = (const float*)d_in[29];

    const int N = in_sizes[0] / 5;
    const int E = in_sizes[1] / 6;
    const int L = in_sizes[15] / HSQ;

    // Workspace layout (floats)
    float* ws   = (float*)d_ws;
    float* ea   = ws;                              // E*128
    float* hA   = ea  + (size_t)E * HDIM;          // N*128
    float* hB   = hA  + (size_t)N * HDIM;          // N*128
    float* agg  = hB  + (size_t)N * HDIM;          // N*128
    float* tbuf = agg + (size_t)N * HDIM;          // N*128 (also heads intermediate)
    float* wcat = tbuf + (size_t)N * HDIM;         // 128*128
    float* bcat = wcat + HSQ;                      // 128

    const dim3 blk(256);
    const int gN = (N + 127) / 128;
    const int gE = (E + 127) / 128;

    pack_heads_kernel<<<(HSQ + HDIM + 255) / 256, blk, 0, stream>>>(
        dh_w1, dh_b1, sh_w1, sh_b1, wcat, bcat);

    encode_kernel<5><<<gN, blk, 0, stream>>>(
        x, ne_w1, ne_b1, ne_w2, ne_b2, ne_g, ne_be, hA, N);
    encode_kernel<6><<<gE, blk, 0, stream>>>(
        edge_attr, ee_w1, ee_b1, ee_w2, ee_b2, ee_g, ee_be, ea, E);

    float* hc = hA;
    float* hn = hB;
    for (int l = 0; l < L; ++l) {
        const long n4 = (long)N * HDIM / 4;
        zero4_kernel<<<(int)((n4 + 255) / 256), blk, 0, stream>>>(agg, n4);
        const long sthreads = (long)E * 32;
        scatter_kernel<<<(int)((sthreads + 255) / 256), blk, 0, stream>>>(
            hc, ea, ei, agg, E);
        gemm128_kernel<true, false><<<gN, blk, 0, stream>>>(
            agg, hc, conv_w1 + (size_t)l * HSQ, conv_b1 + l * HDIM,
            nullptr, nullptr, nullptr, tbuf, N);
        gemm128_kernel<false, true><<<gN, blk, 0, stream>>>(
            tbuf, nullptr, conv_w2 + (size_t)l * HSQ, conv_b2 + l * HDIM,
            hc, pn_g + l * HDIM, pn_be + l * HDIM, hn, N);
        float* tmp = hc; hc = hn; hn = tmp;
    }

    // Heads first layer: relu(h @ [dh_w1|sh_w1] + [dh_b1|sh_b1])
    gemm128_kernel<false, false><<<gN, blk, 0, stream>>>(
        hc, nullptr, wcat, bcat, nullptr, nullptr, nullptr, tbuf, N);

    finalize_kernel<<<(N + 255) / 256, blk, 0, stream>>>(
        tbuf, dh_w2, dh_b2, sh_w2, sh_b2, log_disp, (float*)d_out, N);
}